// dGModule_11699490915126
// MI455X (gfx1250) — compile-verified
//
#include <hip/hip_runtime.h>
#include <hip/hip_bf16.h>
#include <math.h>

typedef float v2f __attribute__((ext_vector_type(2)));
typedef float v8f __attribute__((ext_vector_type(8)));

#define LN_EPS 1e-5f

__device__ inline float gelu_exact(float x) {
    // jax.nn.gelu(approximate=False): 0.5*x*(1+erf(x/sqrt(2)))
    return 0.5f * x * (1.0f + erff(x * 0.70710678118654752f));
}

// Post-process a 16x128 in-register tile held in WMMA f32 C/D layout:
//   acc[t][e] = value at row m = e + 8*(lane>=16), col n = 16*t + (lane&15)
// Applies: +bias, LayerNorm over the 128-wide row (gamma/beta), exact GELU.
__device__ inline void bias_ln_gelu_tile(v8f acc[8],
                                         const float* __restrict__ bias,
                                         const float* __restrict__ gamma,
                                         const float* __restrict__ beta,
                                         int lane) {
    const int nl = lane & 15;
    float bn[8], gn[8], en[8];
#pragma unroll
    for (int t = 0; t < 8; ++t) {
        const int n = t * 16 + nl;
        bn[t] = bias[n];
        gn[t] = gamma[n];
        en[t] = beta[n];
    }
    float s[8], q[8];
#pragma unroll
    for (int e = 0; e < 8; ++e) { s[e] = 0.0f; q[e] = 0.0f; }
#pragma unroll
    for (int t = 0; t < 8; ++t) {
#pragma unroll
        for (int e = 0; e < 8; ++e) {
            const float v = acc[t][e] + bn[t];
            acc[t][e] = v;
            s[e] += v;
            q[e] += v * v;
        }
    }
    // Row data for a given e is spread over the 16 lanes of this half-wave:
    // xor-reduce within groups of 16 (wave32, masks < 16 keep halves apart).
#pragma unroll
    for (int mask = 1; mask <= 8; mask <<= 1) {
#pragma unroll
        for (int e = 0; e < 8; ++e) {
            s[e] += __shfl_xor(s[e], mask, 32);
            q[e] += __shfl_xor(q[e], mask, 32);
        }
    }
    float mu[8], rs[8];
#pragma unroll
    for (int e = 0; e < 8; ++e) {
        mu[e] = s[e] * (1.0f / 128.0f);
        const float var = q[e] * (1.0f / 128.0f) - mu[e] * mu[e];
        rs[e] = rsqrtf(var + LN_EPS);
    }
#pragma unroll
    for (int t = 0; t < 8; ++t) {
#pragma unroll
        for (int e = 0; e < 8; ++e) {
            const float v = (acc[t][e] - mu[e]) * rs[e] * gn[t] + en[t];
            acc[t][e] = gelu_exact(v);
        }
    }
}

// Column-sum a processed tile (over its 16 rows) and atomically accumulate
// scaled sums into dst[128].
__device__ inline void colsum_accum_tile(const v8f acc[8], float* __restrict__ dst,
                                         float scale, int lane) {
#pragma unroll
    for (int t = 0; t < 8; ++t) {
        float cp = 0.0f;
#pragma unroll
        for (int e = 0; e < 8; ++e) cp += acc[t][e];
        cp += __shfl_xor(cp, 16, 32);   // fold rows 0-7 with rows 8-15
        if (lane < 16)
            atomicAdd(&dst[t * 16 + lane], cp * scale);
    }
}

// Pre-pack W[K,128] into WMMA-B fragment order:
//   out[(ks*8+t)*32 + lane] = { W[k][n], W[k+1][n] },
//   k = ks*4 + 2*(lane>=16), n = 16*t + (lane&15).
// Afterwards every B fragment is ONE coalesced 8-byte load.
__global__ void frag_pack_kernel(const float* __restrict__ W, v2f* __restrict__ out) {
    const int idx  = blockIdx.x * 256 + threadIdx.x;   // grid.x == K/4
    const int lane = idx & 31;
    const int t    = (idx >> 5) & 7;
    const int ks   = idx >> 8;
    const int k    = ks * 4 + ((lane >> 4) << 1);
    const int n    = t * 16 + (lane & 15);
    v2f w;
    w.x = W[(size_t)k * 128 + n];
    w.y = W[(size_t)(k + 1) * 128 + n];
    out[idx] = w;
}

// Fused X[R,256] @ W[256,128] + bias -> LayerNorm -> GELU.
// B matrix consumed from pre-packed fragment layout (Wf, 64 ks-steps).
// 8 waves/block, one 16x128 tile per wave, 128 rows/block.
__global__ __launch_bounds__(256) void ff_wmma_kernel(
    const float* __restrict__ X,
    const v2f*   __restrict__ Wf,     // [64*8*32] packed fragments
    const float* __restrict__ bias,
    const float* __restrict__ gamma,
    const float* __restrict__ beta,
    float* __restrict__ store,        // [R,128] or nullptr
    float* __restrict__ accum,        // [B,128] or nullptr
    int rows_per_batch, float accum_scale) {
    const int tid  = threadIdx.x;
    const int wid  = tid >> 5;
    const int lane = tid & 31;
    const int nl   = lane & 15;
    const int kb   = (lane >> 4) * 2;
    const int row0 = blockIdx.x * 128 + wid * 16;

    const float* __restrict__ Arow = X + (size_t)(row0 + nl) * 256;

    v8f acc[8] = {};
    for (int ks = 0; ks < 64; ++ks) {            // K = 256, 4 per WMMA
        const v2f a = *(const v2f*)(Arow + ks * 4 + kb);   // global_load_b64
#pragma unroll
        for (int t = 0; t < 8; ++t) {
            const v2f bf = Wf[(ks * 8 + t) * 32 + lane];   // coalesced b64
            acc[t] = __builtin_amdgcn_wmma_f32_16x16x4_f32(
                false, a, false, bf, (short)0, acc[t], false, false);
        }
    }

    bias_ln_gelu_tile(acc, bias, gamma, beta, lane);

    if (store) {
        const int half = lane >> 4;
#pragma unroll
        for (int t = 0; t < 8; ++t) {
#pragma unroll
            for (int e = 0; e < 8; ++e) {
                store[(size_t)(row0 + e + 8 * half) * 128 + t * 16 + nl] = acc[t][e];
            }
        }
    }
    if (accum) {
        const int batch = row0 / rows_per_batch;  // 16-row tile never straddles
        colsum_accum_tile(acc, accum + batch * 128, accum_scale, lane);
    }
}

// Pairwise stage, fused (never materializes hx):
//   Y_j = hc_r * diag(hc_l[j]) * W_x  -> bias/LN/GELU -> mean accumulate.
// One block per (j, b). LDS holds the fragment-packed, hl-scaled W_x (64 KB);
// each wave keeps TWO 16x128 accumulator tiles so every ds_load_b64 B-fragment
// feeds two WMMAs.
__global__ __launch_bounds__(256) void pair_wmma_kernel(
    const float* __restrict__ hc,     // [B*576, 128]
    const v2f*   __restrict__ Wxf,    // [32*8*32] packed fragments of W_x
    const float* __restrict__ bx,
    const float* __restrict__ gx,
    const float* __restrict__ bex,
    float* __restrict__ c_comb) {     // [B,128] accumulator
    __shared__ v2f Wf[32 * 8 * 32];   // 8192 * 8B = 64 KB

    const int j    = blockIdx.x;      // 0..63
    const int b    = blockIdx.y;      // 0..31
    const int tid  = threadIdx.x;
    const int wid  = tid >> 5;
    const int lane = tid & 31;
    const int nl   = lane & 15;
    const int kb   = (lane >> 4) * 2;

    // Stage diag(hc_l[j]) @ W_x in fragment order: scale the pre-packed W_x
    // fragments by the matching {hl[k], hl[k+1]} pair.
    const float* __restrict__ hl = hc + (size_t)(b * 576 + 512 + j) * 128;
    for (int idx = tid; idx < 32 * 8 * 32; idx += 256) {
        const int li = idx & 31;
        const int ks = idx >> 8;
        const int k  = ks * 4 + ((li >> 4) << 1);
        v2f w = Wxf[idx];
        w.x *= hl[k];
        w.y *= hl[k + 1];
        Wf[idx] = w;
    }
    __syncthreads();

    const float scale = 1.0f / (512.0f * 64.0f);
    for (int mt = 0; mt < 2; ++mt) {              // each wave: 64 rows of i
        const int r0 = wid * 64 + mt * 32;        // two 16-row tiles per pass
        const float* __restrict__ A0 = hc + (size_t)(b * 576 + r0 + nl) * 128;
        const float* __restrict__ A1 = A0 + (size_t)16 * 128;
        v8f acc0[8] = {};
        v8f acc1[8] = {};
        for (int ks = 0; ks < 32; ++ks) {         // K = 128
            const v2f a0 = *(const v2f*)(A0 + ks * 4 + kb);
            const v2f a1 = *(const v2f*)(A1 + ks * 4 + kb);
#pragma unroll
            for (int t = 0; t < 8; ++t) {
                const v2f bf = Wf[(ks * 8 + t) * 32 + lane];  // ds_load_b64
                acc0[t] = __builtin_amdgcn_wmma_f32_16x16x4_f32(
                    false, a0, false, bf, (short)0, acc0[t], false, false);
                acc1[t] = __builtin_amdgcn_wmma_f32_16x16x4_f32(
                    false, a1, false, bf, (short)0, acc1[t], false, false);
            }
        }
        bias_ln_gelu_tile(acc0, bx, gx, bex, lane);
        bias_ln_gelu_tile(acc1, bx, gx, bex, lane);
        colsum_accum_tile(acc0, c_comb + b * 128, scale, lane);
        colsum_accum_tile(acc1, c_comb + b * 128, scale, lane);
    }
}

__global__ void zero_kernel(float* __restrict__ p, int n) {
    const int i = blockIdx.x * blockDim.x + threadIdx.x;
    if (i < n) p[i] = 0.0f;
}

__global__ void finish_kernel(const float* __restrict__ c_comb,
                              const float* __restrict__ c_rec,
                              const float* __restrict__ c_lig,
                              const float* __restrict__ W_dg,
                              const float* __restrict__ b_dg,
                              float* __restrict__ out) {
    const int b = threadIdx.x;
    if (b >= 32) return;
    float s0 = b_dg[0], s1 = b_dg[1];
    for (int c = 0; c < 128; ++c) {
        const float v = c_comb[b * 128 + c] + c_rec[b * 128 + c] + c_lig[b * 128 + c];
        s0 += v * W_dg[c * 2 + 0];
        s1 += v * W_dg[c * 2 + 1];
    }
    const float m  = fmaxf(s0, s1);
    const float e0 = expf(s0 - m), e1 = expf(s1 - m);
    const float inv = 1.0f / (e0 + e1);
    out[b * 2 + 0] = e0 * inv;
    out[b * 2 + 1] = e1 * inv;
}

extern "C" void kernel_launch(void* const* d_in, const int* in_sizes, int n_in,
                              void* d_out, int out_size, void* d_ws, size_t ws_size,
                              hipStream_t stream) {
    (void)in_sizes; (void)n_in; (void)out_size; (void)ws_size;
    const float* h_comb  = (const float*)d_in[0];
    const float* h_rec   = (const float*)d_in[1];
    const float* h_lig   = (const float*)d_in[2];
    const float* W_comb  = (const float*)d_in[3];
    const float* b_comb  = (const float*)d_in[4];
    const float* g_comb  = (const float*)d_in[5];
    const float* be_comb = (const float*)d_in[6];
    const float* W_rec   = (const float*)d_in[7];
    const float* b_rec   = (const float*)d_in[8];
    const float* g_rec   = (const float*)d_in[9];
    const float* be_rec  = (const float*)d_in[10];
    const float* W_x     = (const float*)d_in[11];
    const float* b_x     = (const float*)d_in[12];
    const float* g_x     = (const float*)d_in[13];
    const float* be_x    = (const float*)d_in[14];
    const float* W_dg    = (const float*)d_in[15];
    const float* b_dg    = (const float*)d_in[16];
    float* out = (float*)d_out;

    // Workspace layout (floats):
    //   hc [18432*128] | c_rec[4096] | c_lig[4096] | c_comb[4096]
    //   | Wf_comb [64*8*32 v2f] | Wf_rec [64*8*32 v2f] | Wf_x [32*8*32 v2f]
    float* ws      = (float*)d_ws;
    float* hc      = ws;
    float* c_rec   = hc + (size_t)18432 * 128;
    float* c_lig   = c_rec + 32 * 128;
    float* c_comb  = c_lig + 32 * 128;
    v2f*   Wf_comb = (v2f*)(c_comb + 32 * 128);
    v2f*   Wf_rec  = Wf_comb + 64 * 8 * 32;
    v2f*   Wf_x    = Wf_rec + 64 * 8 * 32;

    // Zero the three [32,128] accumulators (contiguous).
    zero_kernel<<<(3 * 32 * 128 + 255) / 256, 256, 0, stream>>>(c_rec, 3 * 32 * 128);

    // Pre-pack weights into WMMA-B fragment order (one b64 per fragment later).
    frag_pack_kernel<<<64, 256, 0, stream>>>(W_comb, Wf_comb);
    frag_pack_kernel<<<64, 256, 0, stream>>>(W_rec, Wf_rec);
    frag_pack_kernel<<<32, 256, 0, stream>>>(W_x, Wf_x);

    // hc = GELU(LN(h_comb@W_comb + b)): 18432 rows -> 144 blocks, stored to ws.
    ff_wmma_kernel<<<144, 256, 0, stream>>>(h_comb, Wf_comb, b_comb, g_comb, be_comb,
                                            hc, nullptr, 576, 0.0f);
    // hr -> c_rec (mean over nr=512 per batch), activations discarded.
    ff_wmma_kernel<<<128, 256, 0, stream>>>(h_rec, Wf_rec, b_rec, g_rec, be_rec,
                                            nullptr, c_rec, 512, 1.0f / 512.0f);
    // hl -> c_lig (mean over nl=64 per batch).
    ff_wmma_kernel<<<16, 256, 0, stream>>>(h_lig, Wf_rec, b_rec, g_rec, be_rec,
                                           nullptr, c_lig, 64, 1.0f / 64.0f);
    // Fused pairwise stage: 2048 blocks of 512x128x128 GEMM + LN/GELU + mean.
    pair_wmma_kernel<<<dim3(64, 32), 256, 0, stream>>>(hc, Wf_x, b_x, g_x, be_x, c_comb);
    // Head: c_sum @ W_dg + b_dg, softmax.
    finish_kernel<<<1, 32, 0, stream>>>(c_comb, c_rec, c_lig, W_dg, b_dg, out);
}